// DecoderNetwork_Maze_86844238725481
// MI455X (gfx1250) — compile-verified
//
#include <hip/hip_runtime.h>

// ---------------------------------------------------------------------------
// Edge-scoring decoder:  out[e] = sigmoid( x[src[e]]·W_src + x[dst[e]]·W_dst )
// Phase 1: per-node scalars s[i]=x[i]·W_src, d[i]=x[i]·W_dst via fp32 WMMA.
// Phase 2: per-edge gather of two floats from L2-resident arrays + sigmoid.
// ---------------------------------------------------------------------------

typedef __attribute__((ext_vector_type(2))) float v2f;
typedef __attribute__((ext_vector_type(8))) float v8f;

#define D_NODE 128

// One wave (32 lanes) computes s/d for a tile of 16 nodes using
// V_WMMA_F32_16X16X4_F32, accumulating over K=128 in steps of 4.
//
// A (16x4 f32) lane layout (ISA 7.12.2): lanes 0-15 hold M=lane, K={0,1};
// lanes 16-31 hold M=lane-16, K={2,3}.  B (4x16) mirrors the K fold with
// N=lane%16.  Column 0 of B carries W_src, column 1 carries W_dst, rest 0.
// D (16x16 f32): VGPR r, lanes 0-15 -> (M=r, N=lane); lanes 16-31 -> (M=8+r).
// So lane {0,16} hold s for rows {0..7, 8..15}; lanes {1,17} hold d.
__global__ __launch_bounds__(256) void node_score_kernel(
    const float* __restrict__ x, const float* __restrict__ W,
    float* __restrict__ sArr, float* __restrict__ dArr,
    int nNodes, int nTiles)
{
    const int wave = threadIdx.x >> 5;
    const int lane = threadIdx.x & 31;
    const int tile = blockIdx.x * 8 + wave;
    if (tile >= nTiles) return;          // uniform per wave -> EXEC stays all-1s

    const int m    = lane & 15;          // A row / B,D column
    const int half = lane >> 4;          // which K pair this lane holds

    int node = tile * 16 + m;
    if (node >= nNodes) node = nNodes - 1;   // clamp (padding rows recompute a valid row)
    const float* __restrict__ xrow = x + (size_t)node * D_NODE;

    // Column 0 -> W_src = W[0:128], column 1 -> W_dst = W[128:256], others 0.
    const float* __restrict__ wb = W + ((m == 0) ? 0 : D_NODE);
    const bool wact = (m < 2);

    v8f acc = {};
    #pragma unroll
    for (int k0 = 0; k0 < D_NODE; k0 += 4) {
        const int ka = k0 + 2 * half;

        const float2 av = *(const float2*)(xrow + ka);   // 8B aligned: row stride 512B
        v2f a; a.x = av.x; a.y = av.y;

        const float2 wv = *(const float2*)(wb + ka);     // in-bounds for all lanes
        v2f b; b.x = wact ? wv.x : 0.0f;
               b.y = wact ? wv.y : 0.0f;

        // 8 args: (neg_a, A, neg_b, B, c_mod, C, reuse_a, reuse_b)
        acc = __builtin_amdgcn_wmma_f32_16x16x4_f32(
            false, a, false, b, (short)0, acc, false, false);
    }

    // Lanes 0,1,16,17 each hold 8 contiguous results in their accumulator.
    if (m < 2) {
        float* __restrict__ outp = (m == 0 ? sArr : dArr) + tile * 16 + half * 8;
        #pragma unroll
        for (int r = 0; r < 8; ++r) outp[r] = acc[r];
    }
}

// Per-edge: two 4-byte gathers (L2-resident 200KB arrays) + sigmoid.
__global__ __launch_bounds__(256) void edge_sigmoid_kernel(
    const long long* __restrict__ ei,    // int64 edge_index, shape (2, nE) row-major
    const float* __restrict__ sArr, const float* __restrict__ dArr,
    float* __restrict__ out, int nE)
{
    const int e = blockIdx.x * 256 + threadIdx.x;
    if (e >= nE) return;

    const int si = (int)ei[e];                 // row 0: src
    const int di = (int)ei[(size_t)nE + e];    // row 1: dst

    const float z = sArr[si] + dArr[di];
    out[e] = 1.0f / (1.0f + __expf(-z));
}

extern "C" void kernel_launch(void* const* d_in, const int* in_sizes, int n_in,
                              void* d_out, int out_size, void* d_ws, size_t ws_size,
                              hipStream_t stream)
{
    const float*     x  = (const float*)d_in[0];       // (nNodes, 128) fp32
    const long long* ei = (const long long*)d_in[1];   // (2, nE) int64
    const float*     W  = (const float*)d_in[2];       // (1, 256) fp32
    float*           out = (float*)d_out;

    const int nNodes = in_sizes[0] / D_NODE;
    const int nE     = in_sizes[1] / 2;
    const int nTiles = (nNodes + 15) / 16;

    // Workspace: s[nTiles*16] then d[nTiles*16]  (~400 KB for 50k nodes).
    float* sArr = (float*)d_ws;
    float* dArr = sArr + (size_t)nTiles * 16;

    node_score_kernel<<<(nTiles + 7) / 8, 256, 0, stream>>>(
        x, W, sArr, dArr, nNodes, nTiles);
    edge_sigmoid_kernel<<<(nE + 255) / 256, 256, 0, stream>>>(
        ei, sArr, dArr, out, nE);
}